// orderedLossGeo_32581621908068
// MI455X (gfx1250) — compile-verified
//
#include <hip/hip_runtime.h>
#include <hip/hip_bf16.h>
#include <stdint.h>

#define PP 4
#define BB 32
#define NN 2048
#define MM 512
#define DLAT 128
#define BIGF 1e9f
#define BETAF 1e-3f

typedef float v2f __attribute__((ext_vector_type(2)));
typedef float v8f __attribute__((ext_vector_type(8)));

// One workgroup per (part p, batch b). 256 threads = 8 wave32.
// D_tile = A(-2x) * B(y) + C, where C carries |x|^2+rowpen + |y|^2+colpen,
// so the WMMA output is directly the masked squared distance.
__global__ __launch_bounds__(256) void parts_chamfer_kernel(
    const float* __restrict__ ipt, const float* __restrict__ opt,
    const float* __restrict__ trans, const float* __restrict__ rots,
    const float* __restrict__ probs, const int* __restrict__ idx,
    float* __restrict__ ws)
{
    // Packed WMMA fragments: per point {c0,c1,c2,0}; lane reads v2f at +half*8B.
    __shared__ __attribute__((aligned(16))) float xfrag[NN*4];  // -2 * rotated x
    __shared__ __attribute__((aligned(16))) float xxp[NN];      // |x|^2 + (sel?0:BIG)
    __shared__ unsigned char xvalid[NN];
    __shared__ __attribute__((aligned(16))) float yfrag[MM*4];  // y
    __shared__ float yyc[MM];                                   // |y|^2 + (valid?0:BIG)
    __shared__ unsigned char yvalid[MM];
    __shared__ unsigned colminbits[MM];
    __shared__ __attribute__((aligned(16))) float yraw[MM*3];   // async-staged raw y
    __shared__ float sR[9], sT[3];
    __shared__ float s_qual, s_cov;
    __shared__ int s_nx, s_ny;

    const int tid = threadIdx.x;
    const int blk = blockIdx.x;
    const int p = blk / BB;
    const int b = blk % BB;

    // ---- Stage 1: kick off async global->LDS copy of this part's y points ----
    // (ASYNCcnt path; overlaps with the x transform below.)
    {
        const float* gy = opt + (size_t)(p*BB + b)*MM*3;
        unsigned lbase = (unsigned)(uintptr_t)&yraw[0];
        #pragma unroll
        for (int i = 0; i < (MM*3)/256; ++i) {
            int e = tid + i*256;
            unsigned laddr = lbase + (unsigned)(e*4);
            unsigned long long ga = (unsigned long long)(uintptr_t)(gy + e);
            asm volatile("global_load_async_to_lds_b32 %0, %1, off"
                         :: "v"(laddr), "v"(ga) : "memory");
        }
    }

    // ---- Stage 2: init accumulators / rotation matrix / column-min table ----
    if (tid == 0) {
        s_qual = 0.f; s_cov = 0.f; s_nx = 0; s_ny = 0;
        const float* q = rots + (size_t)(p*BB + b)*4;
        float qw=q[0], qx=q[1], qy=q[2], qz=q[3];
        float inv = rsqrtf(qw*qw + qx*qx + qy*qy + qz*qz);
        qw*=inv; qx*=inv; qy*=inv; qz*=inv;
        sR[0]=1.f-2.f*(qy*qy+qz*qz); sR[1]=2.f*(qx*qy-qw*qz); sR[2]=2.f*(qx*qz+qw*qy);
        sR[3]=2.f*(qx*qy+qw*qz);     sR[4]=1.f-2.f*(qx*qx+qz*qz); sR[5]=2.f*(qy*qz-qw*qx);
        sR[6]=2.f*(qx*qz-qw*qy);     sR[7]=2.f*(qy*qz+qw*qx); sR[8]=1.f-2.f*(qx*qx+qy*qy);
        const float* t = trans + (size_t)(p*BB + b)*3;
        sT[0]=t[0]; sT[1]=t[1]; sT[2]=t[2];
    }
    for (int c = tid; c < MM; c += 256) colminbits[c] = __float_as_uint(3e38f);
    __syncthreads();

    // ---- Stage 3: rotate/stage x points (global reads overlap async y copy) ----
    int cntx = 0;
    {
        float R0=sR[0],R1=sR[1],R2=sR[2],R3=sR[3],R4=sR[4],R5=sR[5],R6=sR[6],R7=sR[7],R8=sR[8];
        float t0=sT[0],t1=sT[1],t2=sT[2];
        for (int n = tid; n < NN; n += 256) {
            const float* xp = ipt + ((size_t)(b*NN + n))*3;
            float ax=xp[0], ay=xp[1], az=xp[2];
            bool nz  = (ax!=0.f) || (ay!=0.f) || (az!=0.f);
            bool sel = (idx[b*NN + n] == p) && nz;
            float dx=ax-t0, dy=ay-t1, dz=az-t2;
            float lx = R0*dx + R1*dy + R2*dz;
            float ly = R3*dx + R4*dy + R5*dz;
            float lz = R6*dx + R7*dy + R8*dz;
            *(float4*)&xfrag[n << 2] = make_float4(-2.f*lx, -2.f*ly, -2.f*lz, 0.f);
            xxp[n] = lx*lx + ly*ly + lz*lz + (sel ? 0.f : BIGF);
            xvalid[n] = sel ? 1 : 0;
            cntx += sel ? 1 : 0;
        }
    }
    if (cntx) atomicAdd(&s_nx, cntx);

    // ---- Stage 4: wait for async y copy, then build y fragments from LDS ----
    asm volatile("s_wait_asynccnt 0" ::: "memory");
    __syncthreads();

    int cnty = 0;
    for (int c = tid; c < MM; c += 256) {
        float ax = yraw[c*3+0], ay = yraw[c*3+1], az = yraw[c*3+2];
        bool valid = (ax!=0.f) || (ay!=0.f) || (az!=0.f);
        *(float4*)&yfrag[c << 2] = make_float4(ax, ay, az, 0.f);
        float yy = ax*ax + ay*ay + az*az;
        yyc[c] = valid ? yy : yy + BIGF;
        yvalid[c] = valid ? 1 : 0;
        cnty += valid ? 1 : 0;
    }
    if (cnty) atomicAdd(&s_ny, cnty);
    __syncthreads();

    // ---- Stage 5: WMMA pairwise-distance tiles ----
    const int wave = tid >> 5;
    const int lane = tid & 31;
    const int l15  = lane & 15;
    const int half = lane >> 4;

    float qacc = 0.f;

    for (int t = wave; t < NN/16; t += 8) {
        const int m0 = t << 4;
        const int arow = m0 + l15;
        // A fragment (16x4 f32): lanes 0-15 K=0,1 ; lanes 16-31 K=2,3 (branchless)
        v2f a = *(const v2f*)&xfrag[(arow << 2) + (half << 1)];
        // per-(VGPR j) row bias |x|^2 + rowpen for rows m0+half*8+j
        float4 ea = *(const float4*)&xxp[m0 + (half << 3)];
        float4 eb = *(const float4*)&xxp[m0 + (half << 3) + 4];
        float e[8] = {ea.x, ea.y, ea.z, ea.w, eb.x, eb.y, eb.z, eb.w};
        float rm[8];
        #pragma unroll
        for (int j = 0; j < 8; ++j) rm[j] = 3e38f;

        for (int ct = 0; ct < MM/16; ++ct) {
            const int col = (ct << 4) + l15;
            // B fragment (4x16 f32): lanes 0-15 K=0,1 ; lanes 16-31 K=2,3 (branchless)
            v2f bf = *(const v2f*)&yfrag[(col << 2) + (half << 1)];
            float cc = yyc[col];
            v8f c;
            #pragma unroll
            for (int j = 0; j < 8; ++j) c[j] = cc + e[j];
            v8f acc = __builtin_amdgcn_wmma_f32_16x16x4_f32(
                false, a, false, bf, (short)0, c, false, false);
            #pragma unroll
            for (int j = 0; j < 8; ++j) rm[j] = fminf(rm[j], acc[j]);
            float c0 = fminf(fminf(acc[0], acc[1]), fminf(acc[2], acc[3]));
            float c1 = fminf(fminf(acc[4], acc[5]), fminf(acc[6], acc[7]));
            float cand = fminf(c0, c1);
            // both halves hold the same column; distances >= 0 so uint order == float order
            atomicMin(&colminbits[col], __float_as_uint(fmaxf(cand, 0.f)));
        }
        // row-min reduce across the 16 lanes of each half
        #pragma unroll
        for (int off = 1; off < 16; off <<= 1) {
            #pragma unroll
            for (int j = 0; j < 8; ++j)
                rm[j] = fminf(rm[j], __shfl_xor(rm[j], off, 32));
        }
        if (l15 == 0) {
            #pragma unroll
            for (int j = 0; j < 8; ++j)
                if (xvalid[m0 + (half << 3) + j]) qacc += rm[j];
        }
    }
    if (l15 == 0 && qacc != 0.f) atomicAdd(&s_qual, qacc);
    __syncthreads();

    // ---- Stage 6: coverage sum over valid y columns ----
    float cacc = 0.f;
    for (int c = tid; c < MM; c += 256)
        if (yvalid[c]) cacc += __uint_as_float(colminbits[c]);
    if (cacc != 0.f) atomicAdd(&s_cov, cacc);
    __syncthreads();

    if (tid == 0) {
        int nx = s_nx, ny = s_ny;
        float qual = s_qual / (float)(nx > 1 ? nx : 1);
        float cov  = s_cov  / (float)(ny > 1 ? ny : 1);
        float valid = (nx > 0 && ny > 0) ? 1.f : 0.f;
        float* o = ws + (size_t)(p*BB + b)*4;
        o[0] = qual; o[1] = cov; o[2] = valid; o[3] = probs[p*BB + b];
    }
}

// Single-block finalize: KL reduction + per-part weighted averaging -> 5 scalars.
__global__ __launch_bounds__(256) void finalize_kernel(
    const float* __restrict__ ws, const float* __restrict__ mu,
    const float* __restrict__ logvar, float* __restrict__ out)
{
    __shared__ float red[256];
    __shared__ float s_kld;
    const int tid = threadIdx.x;

    float partial = 0.f;
    for (int i = tid; i < BB*DLAT; i += 256) {
        float m = mu[i], lv = logvar[i];
        partial += 1.f + lv - m*m - expf(lv);
    }
    red[tid] = partial;
    __syncthreads();
    for (int s = 128; s > 0; s >>= 1) {
        if (tid < s) red[tid] += red[tid + s];
        __syncthreads();
    }
    if (tid == 0) s_kld = -0.5f * red[0] / (float)BB;
    __syncthreads();

    if (tid == 0) {
        float kld = s_kld;
        float chamfer = 0.f, cover = 0.f, quality = 0.f, num = 0.f;
        for (int i = 0; i < PP; ++i) {
            float n = 0.f, scd = 0.f, scov = 0.f, squal = 0.f;
            for (int bb = 0; bb < BB; ++bb) {
                const float* e = ws + (size_t)(i*BB + bb)*4;
                float q = e[0], c = e[1], v = e[2], w = e[3];
                if (v > 0.5f) {
                    scd  += w * (q + c);
                    scov += w * c;
                    squal+= w * q;
                    n += 1.f;
                }
            }
            float has = (n > 0.f) ? 1.f : 0.f;
            float inv_n = has / (n > 1.f ? n : 1.f);
            chamfer += scd * inv_n;
            cover   += scov * inv_n;
            quality += squal * inv_n;
            num += has;
        }
        float inv_num = 1.f / (num > 1.f ? num : 1.f);
        float cds = chamfer * inv_num;
        float covs = cover * inv_num;
        float quals = quality * inv_num;
        out[0] = cds + BETAF * kld;
        out[1] = cds;
        out[2] = covs;
        out[3] = quals;
        out[4] = kld;
    }
}

extern "C" void kernel_launch(void* const* d_in, const int* in_sizes, int n_in,
                              void* d_out, int out_size, void* d_ws, size_t ws_size,
                              hipStream_t stream) {
    const float* ipt    = (const float*)d_in[0];
    const float* opt    = (const float*)d_in[1];
    const float* trans  = (const float*)d_in[2];
    const float* rots   = (const float*)d_in[3];
    const float* probs  = (const float*)d_in[4];
    const int*   idx    = (const int*)  d_in[5];
    const float* mu     = (const float*)d_in[6];
    const float* logvar = (const float*)d_in[7];
    float* ws = (float*)d_ws;
    float* out = (float*)d_out;

    parts_chamfer_kernel<<<dim3(PP*BB), dim3(256), 0, stream>>>(
        ipt, opt, trans, rots, probs, idx, ws);
    finalize_kernel<<<dim3(1), dim3(256), 0, stream>>>(ws, mu, logvar, out);
}